// LDAWordGenerator_9998683865751
// MI455X (gfx1250) — compile-verified
//
#include <hip/hip_runtime.h>
#include <math.h>

typedef __attribute__((ext_vector_type(2))) float v2f;
typedef __attribute__((ext_vector_type(8))) float v8f;

#define B_DIM 128
#define T_DIM 128
#define V_DIM 32000
#define NCOLS 64     // columns of V per block
#define S2 80        // LDS pair-row stride in float2 units (64 cols + 16 pad)
                     // 2*S2 mod 64 == 32 -> B-frag reads hit all 64 banks once

// ---------------------------------------------------------------------------
// Kernel 1: theta[b,:] = softmax(inputs[b,:]) written directly in WMMA
// A-fragment order so the main kernel's A load is one coalesced b64 per wave
// per k-step.
// ---------------------------------------------------------------------------
__global__ void theta_softmax_kernel(const float* __restrict__ inp,
                                     float* __restrict__ thetaA) {
  __shared__ float red[T_DIM];
  const int b = blockIdx.x;
  const int t = threadIdx.x;
  float x = inp[b * T_DIM + t];
  red[t] = x;
  __syncthreads();
  for (int s = T_DIM / 2; s > 0; s >>= 1) {
    if (t < s) red[t] = fmaxf(red[t], red[t + s]);
    __syncthreads();
  }
  float m = red[0];
  __syncthreads();
  float e = expf(x - m);
  red[t] = e;
  __syncthreads();
  for (int s = T_DIM / 2; s > 0; s >>= 1) {
    if (t < s) red[t] += red[t + s];
    __syncthreads();
  }
  const float th = e / red[0];
  // swizzled A-fragment store: lanes 0-15 hold K%4 in {0,1}, lanes 16-31 {2,3}
  const int kk = t >> 2;            // k-step
  const int khalf2 = (t >> 1) & 1;
  const int j = t & 1;
  const int lane = khalf2 * 16 + (b & 15);
  thetaA[(((b >> 4) * 32 + kk) * 32 + lane) * 2 + j] = th;
}

// ---------------------------------------------------------------------------
// Kernel 2: lse_w[t] = logsumexp(w[t,:])  (128 rows of 32000; 2nd pass L2-hot)
// ---------------------------------------------------------------------------
__global__ void lse_rows_kernel(const float* __restrict__ w,
                                float* __restrict__ lse) {
  __shared__ float red[256];
  const int t = blockIdx.x;
  const int tid = threadIdx.x;
  const float* row = w + (size_t)t * V_DIM;
  float m = -INFINITY;
  for (int v = tid; v < V_DIM; v += 256) m = fmaxf(m, row[v]);
  red[tid] = m;
  __syncthreads();
  for (int s = 128; s > 0; s >>= 1) {
    if (tid < s) red[tid] = fmaxf(red[tid], red[tid + s]);
    __syncthreads();
  }
  m = red[0];
  __syncthreads();
  float sum = 0.0f;
  for (int v = tid; v < V_DIM; v += 256) sum += expf(row[v] - m);
  red[tid] = sum;
  __syncthreads();
  for (int s = 128; s > 0; s >>= 1) {
    if (tid < s) red[tid] += red[tid + s];
    __syncthreads();
  }
  if (tid == 0) lse[t] = m + logf(red[0]);
}

// ---------------------------------------------------------------------------
// Kernel 3 (fused main): per 128x64 tile of w:
//   raw[t,n] = w[t,v0+n] - lse[t]  -> LDS (K-pair layout), track column max
//   cmax[n]  = max_t raw[t,n]      -> LDS reduction
//   p[t,n]   = exp(raw - cmax[n])  -> in place
//   z[b,v]   = cmax[n] + log(theta @ p)  via V_WMMA_F32_16X16X4_F32
// ---------------------------------------------------------------------------
__global__ void __launch_bounds__(256) lda_wmma_kernel(
    const float* __restrict__ w, const float* __restrict__ thetaA,
    const float* __restrict__ lse, float* __restrict__ out) {
  __shared__ v2f p2[64 * S2];  // 40 KB
  __shared__ float lse_s[T_DIM];
  __shared__ float smax[4][NCOLS];
  __shared__ float cmax_s[NCOLS];
  float* pf = (float*)p2;

  const int tid = threadIdx.x;
  const int v0 = blockIdx.x * NCOLS;

  if (tid < T_DIM) lse_s[tid] = lse[tid];
  __syncthreads();

  // ---- stage raw tile + partial column max ----
  const int col = tid & 63;  // 0..63
  const int r0 = tid >> 6;   // 0..3
  float m = -INFINITY;
#pragma unroll
  for (int i = 0; i < 32; ++i) {
    const int row = r0 + i * 4;
    const float raw = w[(size_t)row * V_DIM + v0 + col] - lse_s[row];
    pf[((row >> 1) * S2 + col) * 2 + (row & 1)] = raw;
    m = fmaxf(m, raw);
  }
  smax[r0][col] = m;
  __syncthreads();

  if (tid < 64) {
    cmax_s[tid] = fmaxf(fmaxf(smax[0][tid], smax[1][tid]),
                        fmaxf(smax[2][tid], smax[3][tid]));
  }
  __syncthreads();

  // ---- exponentiate in place ----
  const float cm_col = cmax_s[col];
#pragma unroll
  for (int i = 0; i < 32; ++i) {
    const int row = r0 + i * 4;
    const int a = ((row >> 1) * S2 + col) * 2 + (row & 1);
    pf[a] = expf(pf[a] - cm_col);
  }
  __syncthreads();

  // ---- WMMA main loop: 4 independent accumulator chains per wave ----
  const int wave = tid >> 5;  // 0..7 -> M tile
  const int lane = tid & 31;
  const int nlane = lane & 15;
  const int kodd = (lane < 16) ? 0 : 1;  // which row of the K pair
  const v2f* ap = (const v2f*)thetaA + wave * 32 * 32 + lane;
  const v2f* bp = &p2[kodd * S2 + nlane];  // induction pointer, += 2*S2/step

  v8f acc[4] = {{}, {}, {}, {}};
#pragma unroll
  for (int kk = 0; kk < 32; ++kk) {
    v2f a = ap[kk * 32];  // coalesced: 32 lanes x 8B contiguous
    v2f b0 = bp[0];
    v2f b1 = bp[16];
    v2f b2 = bp[32];
    v2f b3 = bp[48];
    acc[0] = __builtin_amdgcn_wmma_f32_16x16x4_f32(false, a, false, b0,
                                                   (short)0, acc[0], false, false);
    acc[1] = __builtin_amdgcn_wmma_f32_16x16x4_f32(false, a, false, b1,
                                                   (short)0, acc[1], false, false);
    acc[2] = __builtin_amdgcn_wmma_f32_16x16x4_f32(false, a, false, b2,
                                                   (short)0, acc[2], false, false);
    acc[3] = __builtin_amdgcn_wmma_f32_16x16x4_f32(false, a, false, b3,
                                                   (short)0, acc[3], false, false);
    bp += 2 * S2;
  }

  // ---- epilogue: z = cmax + log(acc); D: VGPR r -> M=r (lanes 0-15) or
  // M=r+8 (lanes 16-31), N = lane&15 ----
  const int mbase = wave * 16 + ((lane < 16) ? 0 : 8);
#pragma unroll
  for (int n = 0; n < 4; ++n) {
    const float cm = cmax_s[n * 16 + nlane];
    const int nglob = v0 + n * 16 + nlane;
#pragma unroll
    for (int r = 0; r < 8; ++r) {
      out[(size_t)(mbase + r) * V_DIM + nglob] = cm + logf(acc[n][r]);
    }
  }
}

// ---------------------------------------------------------------------------
extern "C" void kernel_launch(void* const* d_in, const int* in_sizes, int n_in,
                              void* d_out, int out_size, void* d_ws,
                              size_t ws_size, hipStream_t stream) {
  const float* inputs = (const float*)d_in[0];  // (128, 128)
  const float* w = (const float*)d_in[1];       // (128, 32000)
  float* out = (float*)d_out;                   // (128, 32000)

  char* ws = (char*)d_ws;
  float* thetaA = (float*)ws;         // 128*128 f32 = 64 KB (A-fragment order)
  float* lse = (float*)(ws + 65536);  // 128 f32

  theta_softmax_kernel<<<B_DIM, T_DIM, 0, stream>>>(inputs, thetaA);
  lse_rows_kernel<<<T_DIM, 256, 0, stream>>>(w, lse);
  lda_wmma_kernel<<<V_DIM / NCOLS, 256, 0, stream>>>(w, thetaA, lse, out);
}